// DAS_88089779240977
// MI455X (gfx1250) — compile-verified
//
#include <hip/hip_runtime.h>
#include <stdint.h>

// ---------------------------------------------------------------------------
// Delay-and-sum beamformer for MI455X (gfx1250).
//
// Roofline: 128 MB streaming read of samples_idx dominates (~5.6 us at
// 23.3 TB/s); compute is ~0.2 GFLOP (negligible). rfs (2 MB) stays L2-resident.
// Design: stage rfs channel rows into LDS with GLOBAL_LOAD_ASYNC_TO_LDS_B128
// (ASYNCcnt-tracked, double-buffered, 2 channels per phase = 16 KB contiguous),
// turn the 2 random gathers per pixel-channel into LDS reads, and software-
// pipeline the coalesced position loads so the dominant global stream stays
// in flight across barriers.
// ---------------------------------------------------------------------------

// Problem dims (fixed by the reference setup_inputs()).
constexpr int B_      = 4;
constexpr int NC      = 64;
constexpr int NS      = 2048;                // samples per RF line
constexpr int NZ      = 512;
constexpr int NX      = 256;
constexpr int P       = NZ * NX;             // 131072 pixels
constexpr int THREADS = 256;                 // 8 wave32s
constexpr int PXPT    = 8;                   // pixels per thread
constexpr int TPIX    = THREADS * PXPT;      // 2048 pixels per block
constexpr int BLOCKS_PER_B = P / TPIX;       // 64
constexpr int CPP     = 2;                   // channels per phase (contiguous in rfs)
constexpr int PHASES  = NC / CPP;            // 32
constexpr int CHUNKS  = (CPP * NS * 4) / (THREADS * 16);  // 4 B128 per thread/phase

// Async global->LDS 128-bit load. %0 = LDS byte offset (VGPR),
// %1 = 64-bit global address (VGPR pair). Tracked by ASYNCcnt.
#define ASYNC_LOAD_B128(ldsoff, gptr)                                        \
  asm volatile("global_load_async_to_lds_b128 %0, %1, off"                   \
               :: "v"(ldsoff), "v"(gptr) : "memory")

#if defined(__has_builtin)
#if __has_builtin(__builtin_amdgcn_s_wait_asynccnt)
#define WAIT_ASYNC(n) __builtin_amdgcn_s_wait_asynccnt(n)
#endif
#endif
#ifndef WAIT_ASYNC
#define WAIT_ASYNC(n) asm volatile("s_wait_asynccnt " #n ::: "memory")
#endif

__global__ __launch_bounds__(THREADS)
void DAS_88089779240977_kernel(const float* __restrict__ rfs,
                               const float* __restrict__ samples,
                               const int*   __restrict__ ids,
                               float*       __restrict__ out)
{
    __shared__ float srow[2][CPP * NS];      // 2 x 16KB double buffer

    const int t     = threadIdx.x;
    const int tile  = blockIdx.x;            // 0..63
    const int b     = blockIdx.y;            // 0..3
    const int id    = ids[b];
    const int pbase = tile * TPIX + t * PXPT;

    const float* rfs_b  = rfs + (size_t)b * NC * NS;       // K == 1
    const float* smp_px = samples + (size_t)id * NC * P + pbase;

    // Wave-relative LDS byte offsets of the two phase buffers.
    const uint32_t ldsA = (uint32_t)(uintptr_t)(&srow[0][0]) + (uint32_t)t * 16u;
    const uint32_t ldsB = (uint32_t)(uintptr_t)(&srow[1][0]) + (uint32_t)t * 16u;

    // Prime the pipeline: phase 0 (channels 0..1, 16 KB contiguous) -> buffer 0.
    {
        const float* g = rfs_b + t * 4;
#pragma unroll
        for (int k = 0; k < CHUNKS; ++k)
            ASYNC_LOAD_B128(ldsA + (uint32_t)k * 4096u, g + k * 1024);
    }

    // Preload positions for phase 0 (both channels, 2 x float4 each, coalesced).
    float4 q00 = ((const float4*)(smp_px + (size_t)0 * P))[0];
    float4 q01 = ((const float4*)(smp_px + (size_t)0 * P))[1];
    float4 q10 = ((const float4*)(smp_px + (size_t)1 * P))[0];
    float4 q11 = ((const float4*)(smp_px + (size_t)1 * P))[1];

    float acc[PXPT];
#pragma unroll
    for (int j = 0; j < PXPT; ++j) acc[j] = 0.0f;

    for (int ph = 0; ph < PHASES; ++ph) {
        const int buf = ph & 1;

        // Issue next phase's 16 KB into the other buffer; ASYNCcnt is
        // in-order, so waiting <= CHUNKS guarantees this phase's loads landed.
        if (ph + 1 < PHASES) {
            const float*   g = rfs_b + (size_t)(ph + 1) * CPP * NS + t * 4;
            const uint32_t l = buf ? ldsA : ldsB;
#pragma unroll
            for (int k = 0; k < CHUNKS; ++k)
                ASYNC_LOAD_B128(l + (uint32_t)k * 4096u, g + k * 1024);
            WAIT_ASYNC(4);
        } else {
            WAIT_ASYNC(0);
        }
        __syncthreads();                     // phase data visible to all waves

        // Prefetch next phase's positions now; these VMEM loads stay in
        // flight across the compute + barrier below (latency hidden).
        float4 n00 = q00, n01 = q01, n10 = q10, n11 = q11;
        if (ph + 1 < PHASES) {
            const float* sn = smp_px + (size_t)(ph + 1) * CPP * P;
            n00 = ((const float4*)(sn + (size_t)0 * P))[0];
            n01 = ((const float4*)(sn + (size_t)0 * P))[1];
            n10 = ((const float4*)(sn + (size_t)1 * P))[0];
            n11 = ((const float4*)(sn + (size_t)1 * P))[1];
        }

        // Interpolate both channels of this phase out of LDS.
#pragma unroll
        for (int cc = 0; cc < CPP; ++cc) {
            const float* row = &srow[buf][cc * NS];
            const float4 a = cc ? q10 : q00;
            const float4 bq = cc ? q11 : q01;
            const float pv[PXPT] = {a.x, a.y, a.z, a.w, bq.x, bq.y, bq.z, bq.w};
#pragma unroll
            for (int j = 0; j < PXPT; ++j) {
                float pos = fminf(fmaxf(pv[j], 0.0f), (float)(NS - 1)); // border clamp
                float f   = floorf(pos);
                int   i0  = (int)f;
                float w   = pos - f;
                int   i1  = min(i0 + 1, NS - 1);
                float g0  = row[i0];         // LDS gather
                float g1  = row[i1];         // LDS gather
                acc[j] += fmaf(w, g1 - g0, g0);  // g0 + (g1-g0)*w
            }
        }
        __syncthreads();                     // buffer free for reuse (ph+2)

        q00 = n00; q01 = n01; q10 = n10; q11 = n11;
    }

    // Vectorized output store (K == 1).
    float4* o = (float4*)(out + (size_t)b * P + pbase);
    o[0] = make_float4(acc[0], acc[1], acc[2], acc[3]);
    o[1] = make_float4(acc[4], acc[5], acc[6], acc[7]);
}

extern "C" void kernel_launch(void* const* d_in, const int* in_sizes, int n_in,
                              void* d_out, int out_size, void* d_ws, size_t ws_size,
                              hipStream_t stream) {
    (void)in_sizes; (void)n_in; (void)out_size; (void)d_ws; (void)ws_size;
    const float* rfs     = (const float*)d_in[0];
    const float* samples = (const float*)d_in[1];
    const int*   ids     = (const int*)d_in[2];
    float*       out     = (float*)d_out;

    dim3 grid(BLOCKS_PER_B, B_, 1);
    dim3 block(THREADS, 1, 1);
    DAS_88089779240977_kernel<<<grid, block, 0, stream>>>(rfs, samples, ids, out);
}